// Model_86878598463719
// MI455X (gfx1250) — compile-verified
//
#include <hip/hip_runtime.h>
#include <hip/hip_bf16.h>

#define N_NODES   40000
#define N_EDGES   640000
#define IN_CH     128
#define HID_CH    256
#define EDGE_DIM  64
#define N_SAMP    2048
#define N_SLOTS   (1 + 2*N_SAMP)       // 4097 sampled rows (center + pos + neg)
#define EPS_NORM  1e-12f

typedef float v2f __attribute__((ext_vector_type(2)));
typedef float v8f __attribute__((ext_vector_type(8)));

// ---------------- workspace layout (bytes) ----------------
#define WS_AGG    0                        // 40000*128*4      = 20,480,000
#define WS_FLAGS  20480000                 // 40000*4          = 160,000
#define WS_NLIST  20640000                 // 4097*4 (pad)     = 16,640
#define WS_ELIST  20656640                 // 640000*4         = 2,560,000
#define WS_ECNT   23216640                 // 64
#define WS_SUMS   23216704                 // 64
#define WS_WEP    23216768                 // 32,768
#define WS_W1P    23249536                 // 131,072
#define WS_W2P    23380608                 // 262,144
#define WS_EMB    23642752                 // 4097*256*4       = 4,195,328

__device__ __forceinline__ v8f wmma4(v2f a, v2f b, v8f c) {
  return __builtin_amdgcn_wmma_f32_16x16x4_f32(false, a, false, b, (short)0, c,
                                               false, false);
}

// ---------------- 1) pack weights into WMMA-B pair layout ----------------
// layout: pair index (s*2+hi)*N + n  -> (W[4s+2hi][n], W[4s+2hi+1][n])
__global__ void pack_kernel(const float* __restrict__ We,
                            const float* __restrict__ W1,
                            const float* __restrict__ W2,
                            float* __restrict__ Wep,
                            float* __restrict__ W1p,
                            float* __restrict__ W2p) {
  int i = blockIdx.x * blockDim.x + threadIdx.x;
  if (i < 16 * 2 * 128) {                       // We: 64x128, 16 ksteps
    int n = i & 127, hi = (i >> 7) & 1, s = i >> 8;
    int k = 4 * s + 2 * hi;
    Wep[i * 2 + 0] = We[k * IN_CH + n];
    Wep[i * 2 + 1] = We[(k + 1) * IN_CH + n];
  }
  if (i < 32 * 2 * 256) {                       // W1: 128x256, 32 ksteps
    int n = i & 255, hi = (i >> 8) & 1, s = i >> 9;
    int k = 4 * s + 2 * hi;
    W1p[i * 2 + 0] = W1[k * HID_CH + n];
    W1p[i * 2 + 1] = W1[(k + 1) * HID_CH + n];
  }
  if (i < 64 * 2 * 256) {                       // W2: 256x256, 64 ksteps
    int n = i & 255, hi = (i >> 8) & 1, s = i >> 9;
    int k = 4 * s + 2 * hi;
    W2p[i * 2 + 0] = W2[k * HID_CH + n];
    W2p[i * 2 + 1] = W2[(k + 1) * HID_CH + n];
  }
}

// ---------------- 2) zero agg / flags / counters ----------------
__global__ void zero_kernel(float4* __restrict__ agg4, int* __restrict__ flags,
                            unsigned* __restrict__ ctr, float* __restrict__ sums) {
  int i = blockIdx.x * blockDim.x + threadIdx.x;
  int stride = gridDim.x * blockDim.x;
  float4 z = {0.f, 0.f, 0.f, 0.f};
  for (int j = i; j < N_NODES * IN_CH / 4; j += stride) agg4[j] = z;
  for (int j = i; j < N_NODES; j += stride) flags[j] = 0;
  if (i < 8) { ctr[i] = 0u; sums[i] = 0.f; }
}

// ---------------- 3) flag sampled nodes, build slot->node list ----------------
__global__ void setflags_kernel(const int* __restrict__ pos,
                                const int* __restrict__ neg,
                                int* __restrict__ flags,
                                int* __restrict__ nodelist) {
  int i = blockIdx.x * blockDim.x + threadIdx.x;
  if (i == 0) { flags[0] = 1; nodelist[0] = 0; }
  if (i < N_SAMP) {
    int p = pos[i]; flags[p] = 1; nodelist[1 + i] = p;
    int q = neg[i]; flags[q] = 1; nodelist[1 + N_SAMP + i] = q;
  }
}

// ---------------- 4) compact edges whose dst is sampled ----------------
__global__ void compact_kernel(const int* __restrict__ ei,
                               const int* __restrict__ flags,
                               int* __restrict__ elist, int* __restrict__ ecnt) {
  int e = blockIdx.x * blockDim.x + threadIdx.x;
  int pred = 0;
  if (e < N_EDGES) pred = flags[ei[N_EDGES + e]];   // row1 = dst
  unsigned m32 = (unsigned)__ballot(pred);
  int lane = threadIdx.x & 31;
  if (m32) {
    int leader = __ffs(m32) - 1;
    int base = 0;
    if (lane == leader) base = atomicAdd(ecnt, __popc(m32));
    base = __shfl(base, leader, 32);
    if (pred) elist[base + __popc(m32 & ((1u << lane) - 1))] = e;
  }
}

// ---------------- 5) edge GEMM + fused relu + scatter-add ----------------
// per wave: 16 edges x (64->128) via 8 ntiles * 16 wmma_f32_16x16x4
__global__ void edge_kernel(const float* __restrict__ x,
                            const int* __restrict__ ei,
                            const float* __restrict__ edge_attr,
                            const float* __restrict__ be,
                            const float* __restrict__ Wep,
                            const int* __restrict__ elist,
                            const int* __restrict__ ecnt,
                            float* __restrict__ agg) {
  __shared__ float ldsB[16 * 2 * 128 * 2];        // 32 KB packed We
  for (int i = threadIdx.x; i < 16 * 2 * 128 * 2; i += blockDim.x)
    ldsB[i] = Wep[i];
  __syncthreads();

  int count = *ecnt;
  int ntiles = (count + 15) >> 4;
  int lane = threadIdx.x & 31;
  int nl = lane & 15, hi = lane >> 4;
  int gw = blockIdx.x * (blockDim.x >> 5) + (threadIdx.x >> 5);
  int nw = gridDim.x * (blockDim.x >> 5);

  for (int tile = gw; tile < ntiles; tile += nw) {
    if (tile + nw < ntiles)
      __builtin_prefetch(&elist[(tile + nw) * 16 + nl], 0, 0);
    int idx = tile * 16 + nl;
    int valid = (idx < count) ? 1 : 0;
    int el0 = elist[tile * 16];
    int eid = valid ? elist[idx] : el0;
    int src = ei[eid];
    int dst = ei[N_EDGES + eid];

    // A tile: 16 edges x 64 attrs, per-lane 16 float2 (WMMA A layout)
    v2f a[16];
    const float* ap = edge_attr + (long)eid * EDGE_DIM + 2 * hi;
#pragma unroll
    for (int s = 0; s < 16; ++s) a[s] = *(const v2f*)(ap + 4 * s);

    for (int nt = 0; nt < 8; ++nt) {
      v8f acc = {};
#pragma unroll
      for (int s = 0; s < 16; ++s) {
        v2f b = *(const v2f*)(&ldsB[((s * 2 + hi) * 128 + nt * 16 + nl) * 2]);
        acc = wmma4(a[s], b, acc);
      }
      int n = nt * 16 + nl;
      float bias = be[n];
#pragma unroll
      for (int v = 0; v < 8; ++v) {
        int m = v + 8 * hi;                       // output row in tile
        int vm = __shfl(valid, m, 32);
        int sv = __shfl(src, m, 32);
        int dv = __shfl(dst, m, 32);
        float msg = fmaxf(acc[v] + bias + x[sv * IN_CH + n], 0.f);
        if (vm) atomicAdd(&agg[dv * IN_CH + n], msg);
      }
    }
  }
}

// ---------------- 6) node MLP on sampled rows (two fused GEMMs) ----------------
__global__ void mlp_kernel(const float* __restrict__ x,
                           const float* __restrict__ agg,
                           const int* __restrict__ nodelist,
                           const float* __restrict__ W1p,
                           const float* __restrict__ b1,
                           const float* __restrict__ W2p,
                           const float* __restrict__ b2,
                           float* __restrict__ emb) {
  __shared__ float ldsT[2][16 * HID_CH];          // 16x256 f32 per wave (32 KB)
  int lane = threadIdx.x & 31;
  int wib = threadIdx.x >> 5;
  float* tbuf = ldsT[wib];
  int nl = lane & 15, hi = lane >> 4;
  int gw = blockIdx.x * 2 + wib;
  int nw = gridDim.x * 2;
  const int tiles = (N_SLOTS + 15) / 16;          // 257

  for (int tile = gw; tile < tiles; tile += nw) {
    int slot = tile * 16 + nl;
    int valid = (slot < N_SLOTS) ? 1 : 0;
    int node = nodelist[valid ? slot : 0];

    // A = h = x + agg, WMMA A layout, K=128 -> 32 ksteps
    v2f a1[32];
    const float* xp = x + node * IN_CH + 2 * hi;
    const float* gp = agg + node * IN_CH + 2 * hi;
#pragma unroll
    for (int s = 0; s < 32; ++s) {
      v2f xv = *(const v2f*)(xp + 4 * s);
      v2f gv = *(const v2f*)(gp + 4 * s);
      a1[s] = xv + gv;
    }
    // GEMM1: 16x128 @ 128x256 -> relu -> LDS (row-major for transpose re-read)
    for (int nt = 0; nt < 16; ++nt) {
      v8f acc = {};
#pragma unroll
      for (int s = 0; s < 32; ++s) {
        v2f b = *(const v2f*)(W1p + ((s * 2 + hi) * HID_CH + nt * 16 + nl) * 2);
        acc = wmma4(a1[s], b, acc);
      }
      int n = nt * 16 + nl;
      float bias = b1[n];
#pragma unroll
      for (int v = 0; v < 8; ++v)
        tbuf[(v + 8 * hi) * HID_CH + n] = fmaxf(acc[v] + bias, 0.f);
    }
    // GEMM2: 16x256 @ 256x256, A re-read from LDS in WMMA-A layout
    for (int nt = 0; nt < 16; ++nt) {
      v8f acc = {};
#pragma unroll
      for (int s = 0; s < 64; ++s) {
        v2f av = *(const v2f*)(&tbuf[nl * HID_CH + 4 * s + 2 * hi]);
        v2f b = *(const v2f*)(W2p + ((s * 2 + hi) * HID_CH + nt * 16 + nl) * 2);
        acc = wmma4(av, b, acc);
      }
      int n = nt * 16 + nl;
      float bias = b2[n];
#pragma unroll
      for (int v = 0; v < 8; ++v) {
        int m = v + 8 * hi;
        int vm = __shfl(valid, m, 32);
        if (vm) emb[(tile * 16 + m) * HID_CH + n] = acc[v] + bias;
      }
    }
  }
}

// ---------------- 7) cosine scores vs center, accumulate sums ----------------
__global__ void dot_kernel(const float* __restrict__ emb,
                           float* __restrict__ sums) {
  int lane = threadIdx.x & 31;
  int w = (blockIdx.x * blockDim.x + threadIdx.x) >> 5;
  if (w >= 2 * N_SAMP) return;
  const float* c = emb;                           // slot 0 = center (raw)
  const float* r = emb + (1 + w) * HID_CH;
  float cc = 0.f, rr = 0.f, cr = 0.f;
  for (int j = lane; j < HID_CH; j += 32) {
    float cv = c[j], rv = r[j];
    cc += cv * cv; rr += rv * rv; cr += cv * rv;
  }
#pragma unroll
  for (int o = 16; o; o >>= 1) {
    cc += __shfl_xor(cc, o, 32);
    rr += __shfl_xor(rr, o, 32);
    cr += __shfl_xor(cr, o, 32);
  }
  if (lane == 0) {
    float score = cr / (fmaxf(sqrtf(cc), EPS_NORM) * fmaxf(sqrtf(rr), EPS_NORM));
    atomicAdd(&sums[(w < N_SAMP) ? 0 : 1], score);
  }
}

// ---------------- 8) final scalar loss ----------------
__global__ void final_kernel(const float* __restrict__ emb,
                             const float* __restrict__ thr_node,
                             const float* __restrict__ sums,
                             float* __restrict__ out) {
  int lane = threadIdx.x & 31;
  float cc = 0.f, tt = 0.f, ct = 0.f;
  for (int j = lane; j < HID_CH; j += 32) {
    float cv = emb[j], tv = thr_node[j];
    cc += cv * cv; tt += tv * tv; ct += cv * tv;
  }
#pragma unroll
  for (int o = 16; o; o >>= 1) {
    cc += __shfl_xor(cc, o, 32);
    tt += __shfl_xor(tt, o, 32);
    ct += __shfl_xor(ct, o, 32);
  }
  if (lane == 0) {
    float thr = ct / (fmaxf(sqrtf(cc), EPS_NORM) * fmaxf(sqrtf(tt), EPS_NORM));
    float pm = sums[0] / (float)N_SAMP;
    float nm = sums[1] / (float)N_SAMP;
    float s1 = 1.f / (1.f + expf(-(pm - thr)));   // T = 1
    float s2 = 1.f / (1.f + expf(-(thr - nm)));
    out[0] = -logf(fmaxf(s1, 1e-12f)) - logf(fmaxf(s2, 1e-12f));
  }
}

extern "C" void kernel_launch(void* const* d_in, const int* in_sizes, int n_in,
                              void* d_out, int out_size, void* d_ws, size_t ws_size,
                              hipStream_t stream) {
  const float* x         = (const float*)d_in[0];
  const int*   ei        = (const int*)d_in[1];
  const float* edge_attr = (const float*)d_in[2];
  const int*   pos_ind   = (const int*)d_in[3];
  const int*   neg_ind   = (const int*)d_in[4];
  const float* We        = (const float*)d_in[5];
  const float* be        = (const float*)d_in[6];
  const float* W1        = (const float*)d_in[7];
  const float* b1        = (const float*)d_in[8];
  const float* W2        = (const float*)d_in[9];
  const float* b2        = (const float*)d_in[10];
  const float* thr_node  = (const float*)d_in[11];
  float* out = (float*)d_out;

  char* w = (char*)d_ws;
  float* agg      = (float*)(w + WS_AGG);
  int*   flags    = (int*)(w + WS_FLAGS);
  int*   nodelist = (int*)(w + WS_NLIST);
  int*   elist    = (int*)(w + WS_ELIST);
  int*   ecnt     = (int*)(w + WS_ECNT);
  float* sums     = (float*)(w + WS_SUMS);
  float* Wep      = (float*)(w + WS_WEP);
  float* W1p      = (float*)(w + WS_W1P);
  float* W2p      = (float*)(w + WS_W2P);
  float* emb      = (float*)(w + WS_EMB);

  pack_kernel<<<(64 * 2 * 256 + 255) / 256, 256, 0, stream>>>(We, W1, W2, Wep, W1p, W2p);
  zero_kernel<<<1024, 256, 0, stream>>>((float4*)agg, flags, (unsigned*)ecnt, sums);
  setflags_kernel<<<(N_SAMP + 255) / 256, 256, 0, stream>>>(pos_ind, neg_ind, flags, nodelist);
  compact_kernel<<<(N_EDGES + 255) / 256, 256, 0, stream>>>(ei, flags, elist, ecnt);
  edge_kernel<<<512, 256, 0, stream>>>(x, ei, edge_attr, be, Wep, elist, ecnt, agg);
  mlp_kernel<<<129, 64, 0, stream>>>(x, agg, nodelist, W1p, b1, W2p, b2, emb);
  dot_kernel<<<(2 * N_SAMP * 32 + 255) / 256, 256, 0, stream>>>(emb, sums);
  final_kernel<<<1, 32, 0, stream>>>(emb, thr_node, sums, out);
}